// BboxSemanticAtt_75239237091987
// MI455X (gfx1250) — compile-verified
//
#include <hip/hip_runtime.h>
#include <stdint.h>

// BboxSemanticAtt on MI455X (gfx1250, wave32).
//
// out[b,y,x] = sigmoid( sum_n c_n * [x1<=x<x2] * [y1<=y<y2] ), F=512.
// Bandwidth-bound: 64 MiB output => ~2.9us floor @ 23.3 TB/s. No global
// intermediates: per-row LDS delta scatter + WMMA-based 512-wide prefix scan.
//
// Prefix scan via V_WMMA_F32_16X16X4_F32: D = L16 * X (+C), L16 = lower-
// triangular ones fed as four 16x4 A-panels, X = 16 segments of 16 row
// elements fed as four 4x16 B-panels. Cross-segment offsets from a 16-lane
// shuffle scan of D[15][*] (lanes 16..31, VGPR7 per ISA C/D layout).

#define F    512
#define NBOX 256

typedef __attribute__((ext_vector_type(2))) float v2f;
typedef __attribute__((ext_vector_type(8))) float v8f;

__launch_bounds__(256, 2)
__global__ void bbox_att_kernel(const float* __restrict__ preds,
                                float* __restrict__ out)
{
    __shared__ float raw[NBOX * 5];      // 5120 B: staged via async-to-LDS
    __shared__ int   boxes[NBOX][4];     // 4096 B: {c bits, x1, x2, y1|y2<<16}
    __shared__ float line[8][F];         // 16 KB: one 512-float row per wave

    const int batch = blockIdx.y;
    const int t     = threadIdx.x;
    const int lane  = t & 31;
    const int wid   = t >> 5;

    // ---------- Phase 1: async DMA of this batch's 256x5 preds into LDS ----
    {
        const uint64_t gbase =
            (uint64_t)(uintptr_t)(preds + (size_t)batch * NBOX * 5);
        // Generic pointers to LDS carry the LDS byte offset in the low 32 bits.
        const uint32_t ldsBase = (uint32_t)(uintptr_t)(void*)&raw[0];
        for (int c = t; c < (NBOX * 5 * 4) / 16; c += 256) {   // 320 x 16B
            uint32_t ldst = ldsBase + (uint32_t)c * 16u;
            uint64_t ga   = gbase + (uint64_t)c * 16u;
            asm volatile("global_load_async_to_lds_b128 %0, %1, off"
                         :: "v"(ldst), "v"(ga) : "memory");
        }
        asm volatile("s_wait_asynccnt 0" ::: "memory");
    }
    __syncthreads();

    // ---------- Phase 2: clip/pack boxes (raw layout: conf,x1,y1,x2,y2) ----
    if (t < NBOX) {
        float c   = raw[t * 5 + 0];
        int x1 = min(max((int)floorf(raw[t * 5 + 1] * 512.f), 0), 512);
        int y1 = min(max((int)floorf(raw[t * 5 + 2] * 512.f), 0), 512);
        int x2 = min(max((int)floorf(raw[t * 5 + 3] * 512.f), 0), 512);
        int y2 = min(max((int)floorf(raw[t * 5 + 4] * 512.f), 0), 512);
        if (!((x2 > x1) && (y2 > y1))) c = 0.f;
        boxes[t][0] = __float_as_int(c);
        boxes[t][1] = x1;
        boxes[t][2] = x2;
        boxes[t][3] = y1 | (y2 << 16);
    }
    __syncthreads();

    // ---------- Phase 3: per-wave rows: scatter -> WMMA scan -> sigmoid ----
    // Triangular A-panels, constant per lane (ISA A 16x4 layout:
    // lanes 0-15 hold K=0,1 in VGPR0/1; lanes 16-31 hold K=2,3).
    const int rowI = lane & 15;
    const int ksel = (lane >> 4) << 1;           // 0 | 2
    v2f A[4];
#pragma unroll
    for (int p = 0; p < 4; ++p) {
        const int k0 = 4 * p + ksel;
        A[p].x = (k0     <= rowI) ? 1.f : 0.f;
        A[p].y = (k0 + 1 <= rowI) ? 1.f : 0.f;
    }

    float* myline = line[wid];
    const int y0 = blockIdx.x * 32;

    for (int r = 0; r < 4; ++r) {
        const int y = y0 + r * 8 + wid;

        // Zero this wave's row-delta line (512 floats, 16/lane).
        const float4 z = {0.f, 0.f, 0.f, 0.f};
#pragma unroll
        for (int q = 0; q < 4; ++q)
            *(float4*)&myline[lane * 16 + q * 4] = z;
        asm volatile("" ::: "memory");   // LDS is in-order within a wave

        // Scatter: active boxes add +c@x1, -c@x2 (x==512 never read: drop).
        for (int i = lane; i < NBOX; i += 32) {
            const int4 bx = *(const int4*)boxes[i];
            const float c = __int_as_float(bx.x);
            const int y1 = bx.w & 0xffff, y2 = bx.w >> 16;
            if (c != 0.f && y >= y1 && y < y2) {
                atomicAdd(&myline[bx.y], c);
                if (bx.z < F) atomicAdd(&myline[bx.z], -c);
            }
        }
        asm volatile("" ::: "memory");

        // Scan 512 = 2 chunks of 256 (16 segs x 16).  X[k][j] = elem j*16+k.
        float rowCarry = 0.f;
        float* outRow = out + ((size_t)(batch * F + y)) * (size_t)F;
#pragma unroll
        for (int cc = 0; cc < 2; ++cc) {
            v8f acc = {0.f, 0.f, 0.f, 0.f, 0.f, 0.f, 0.f, 0.f};
            const int segBase = cc * 256 + rowI * 16 + ksel;
#pragma unroll
            for (int p = 0; p < 4; ++p) {
                // B 4x16 panel p: lanes 0-15 hold rows K=4p{+0,+1},
                // lanes 16-31 rows K=4p{+2,+3}  ->  two consecutive elems.
                const v2f b = *(const v2f*)&myline[segBase + 4 * p];
                acc = __builtin_amdgcn_wmma_f32_16x16x4_f32(
                          false, A[p], false, b, (short)0, acc, false, false);
            }
            // Segment totals T[j] = D[15][j] live in lanes 16..31, VGPR7.
            const float tj = __shfl(acc[7], 16 + rowI, 32);
            float sc = tj;                       // 16-wide inclusive scan
#pragma unroll
            for (int o = 1; o < 16; o <<= 1) {
                const float up = __shfl_up(sc, o, 16);
                if (rowI >= o) sc += up;
            }
            const float offs = (sc - tj) + rowCarry;    // exclusive + carry
            rowCarry += __shfl(sc, 15, 16);             // chunk total

            // Lane l<16 holds elems rowI*16+0..7; lanes>=16 hold +8..15.
            const int xb = cc * 256 + rowI * 16 + (ksel << 2);
            float s[8];
#pragma unroll
            for (int v = 0; v < 8; ++v) {
                const float val = acc[v] + offs;
                s[v] = __builtin_amdgcn_rcpf(1.f + __expf(-val));  // sigmoid
            }
            const float4 lo = {s[0], s[1], s[2], s[3]};
            const float4 hi = {s[4], s[5], s[6], s[7]};
            *(float4*)&outRow[xb]     = lo;
            *(float4*)&outRow[xb + 4] = hi;
        }
    }
}

extern "C" void kernel_launch(void* const* d_in, const int* in_sizes, int n_in,
                              void* d_out, int out_size, void* d_ws, size_t ws_size,
                              hipStream_t stream)
{
    (void)in_sizes; (void)n_in; (void)d_ws; (void)ws_size; (void)out_size;
    const float* preds = (const float*)d_in[0];
    float* out = (float*)d_out;
    // 64 batches x 16 row-groups (32 rows each) = 1024 blocks of 256 threads.
    dim3 grid(16, 64);
    bbox_att_kernel<<<grid, dim3(256), 0, stream>>>(preds, out);
}